// _BinaryConv2d_38946763440195
// MI455X (gfx1250) — compile-verified
//
#include <hip/hip_runtime.h>

typedef __attribute__((ext_vector_type(2))) float v2f;
typedef __attribute__((ext_vector_type(8))) float v8f;

#define IN_H   1024
#define IN_W   1024
#define C_IN   16
#define C_OUT  16
#define OUT_H  512
#define OUT_W  512
#define NBATCH 8
#define TILE_WO 16
#define TILE_HO 8
#define LDS_H  17           // TILE_HO*2 + 1 input rows
#define LDS_W  36           // TILE_WO*2 + 1 = 33 cols, padded to 36
#define CPITCH 17           // channel pitch (odd -> conflict-free LDS banks)
#define KTOT   144          // 9 * 16, K order: k' = (kh*3+kw)*16 + ci
#define KPAD   146          // weight row pitch (even: 8B-aligned v2f loads)

__global__ __launch_bounds__(256)
void binconv2d_wmma_f32(const float* __restrict__ x,
                        const float* __restrict__ w,
                        float* __restrict__ out) {
  __shared__ float lds_w[C_OUT * KPAD];            // decoded +/-1 weights, [co][k']
  __shared__ float lds_x[LDS_H * LDS_W * CPITCH];  // input tile, [h][wc][ci]

  const int tid     = threadIdx.x;
  const int wo_base = blockIdx.x * TILE_WO;
  const int ho_base = blockIdx.y * TILE_HO;
  const int n       = blockIdx.z;

  // ---- stage decoded weights, remapped to k' = (k%9)*16 + (k/9) ----
  for (int i = tid; i < C_OUT * KTOT; i += 256) {
    const int co = i / KTOT;
    const int k  = i - co * KTOT;              // ci*9 + kh*3 + kw (memory order)
    const float v = w[i];
    lds_w[co * KPAD + (k % 9) * 16 + (k / 9)] = (v == 1.0f) ? 1.0f : -1.0f;
  }

  // ---- stage input tile, channel-last, zero padded ----
  // logical loop (ci, h, wc) with wc fastest -> coalesced global reads;
  // LDS store index = (h*LDS_W + wc)*CPITCH + ci, lane stride 17 -> no conflicts
  const float* xn = x + (size_t)n * C_IN * IN_H * IN_W;
  for (int i = tid; i < C_IN * LDS_H * LDS_W; i += 256) {
    const int ci  = i / (LDS_H * LDS_W);
    const int rem = i - ci * (LDS_H * LDS_W);
    const int h   = rem / LDS_W;
    const int wc  = rem - h * LDS_W;
    const int hi  = ho_base * 2 - 1 + h;
    const int wi  = wo_base * 2 - 1 + wc;
    float v = 0.0f;
    if (wc < 33 && hi >= 0 && hi < IN_H && wi >= 0 && wi < IN_W)
      v = xn[(size_t)ci * IN_H * IN_W + (size_t)hi * IN_W + wi];
    lds_x[rem * CPITCH + ci] = v;
  }
  __syncthreads();

  // ---- implicit GEMM: D[co 16][pix 16] += A[co][k'] * B[k'][pix], 36 steps of K=4 ----
  const int lane = tid & 31;
  const int wave = tid >> 5;        // ho_local 0..7 (one output row per wave)
  const int m    = lane & 15;       // A: row M=co ; B: col N=pixel
  const int half = lane >> 4;       // 0 -> K pair {k0,k0+1}; 1 -> {k0+2,k0+3}
  const int kb   = half * 2;

  const float* base_a = lds_w + m * KPAD + kb;
  const float* base_b = lds_x + (wave * 2 * LDS_W + m * 2) * CPITCH + kb;

  v8f acc = {};
#pragma unroll
  for (int g = 0; g < KTOT / 4; ++g) {
    const int k0 = g * 4;
    const int rk = k0 >> 4;                       // kh*3 + kw  (0..8)
    const int c0 = k0 & 15;                       // ci group base (0,4,8,12)
    const int bo = ((rk / 3) * LDS_W + (rk % 3)) * CPITCH + c0;  // compile-time const
    const v2f a = *(const v2f*)(base_a + k0);     // one 8B LDS load, imm offset
    v2f b;
    b.x = base_b[bo];                             // ds_load_b32, imm offset
    b.y = base_b[bo + 1];
    // (neg_a, A, neg_b, B, c_mod, C, reuse_a, reuse_b)
    acc = __builtin_amdgcn_wmma_f32_16x16x4_f32(false, a, false, b,
                                                (short)0, acc, false, false);
  }

  // ---- store D: lane holds pixel m; VGPR r holds co = r + half*8 ----
  const int ho  = ho_base + wave;
  const int wo  = wo_base + m;
  const int co0 = half * 8;
  float* op = out + ((size_t)n * C_OUT + co0) * OUT_H * OUT_W
                  + (size_t)ho * OUT_W + wo;
#pragma unroll
  for (int r = 0; r < 8; ++r) {
    op[(size_t)r * OUT_H * OUT_W] = acc[r];
  }
}

extern "C" void kernel_launch(void* const* d_in, const int* in_sizes, int n_in,
                              void* d_out, int out_size, void* d_ws, size_t ws_size,
                              hipStream_t stream) {
  (void)in_sizes; (void)n_in; (void)d_ws; (void)ws_size; (void)out_size;
  const float* x = (const float*)d_in[0];
  const float* w = (const float*)d_in[1];
  float* out = (float*)d_out;
  dim3 grid(OUT_W / TILE_WO, OUT_H / TILE_HO, NBATCH);  // 32 x 64 x 8 = 16384 blocks
  binconv2d_wmma_f32<<<grid, 256, 0, stream>>>(x, w, out);
}